// EFDMix_19791209300609
// MI455X (gfx1250) — compile-verified
//
#include <hip/hip_runtime.h>
#include <hip/hip_bf16.h>
#include <stdint.h>

// EFDMix on MI455X (gfx1250):
//   per (b,c) row of 4096 floats:
//     sort own row as (sortable_bits<<32 | index)  -> stable argsort
//     sort permuted-batch row values
//     out[i] = x[i] + (sortedPerm[rank(i)] - x[i]) * (1 - lambda[b])
// One workgroup per row. Rows staged into LDS with gfx1250 async
// global->LDS copies, bitonic sort in LDS, and the output row pushed
// back with async LDS->global stores (both directions on ASYNCcnt).

#define B_DIM 32
#define C_DIM 256
#define N_ELEM 4096
#define NT 256

__device__ __forceinline__ unsigned b2k(unsigned u) {
  // monotone float-bits -> unsigned key (total order)
  return (u & 0x80000000u) ? ~u : (u | 0x80000000u);
}
__device__ __forceinline__ float k2f(unsigned k) {
  unsigned u = (k & 0x80000000u) ? (k & 0x7FFFFFFFu) : ~k;
  return __uint_as_float(u);
}

__global__ __launch_bounds__(NT) void efdmix_kernel(
    const float* __restrict__ x, const float* __restrict__ lmda,
    const long long* __restrict__ perm, float* __restrict__ out) {
  __shared__ __align__(16) unsigned long long sA[N_ELEM]; // 32KB own-row composite keys
  __shared__ __align__(16) unsigned sB[N_ELEM];           // 16KB perm-row keys
  __shared__ __align__(16) float sOut[N_ELEM];            // 16KB staging (raw floats, then output)

  const int row = blockIdx.x;            // b*C + c
  const int b   = row / C_DIM;
  const int c   = row - b * C_DIM;
  const int tid = threadIdx.x;
  const int pb  = (int)perm[b];
  const float lm = lmda[b];

  const unsigned rowABytes = (unsigned)row * (N_ELEM * 4u);
  const unsigned rowBBytes = (unsigned)(pb * C_DIM + c) * (N_ELEM * 4u);

  // LDS byte addresses (flat-pointer low 32 bits == LDS offset on gfx1250)
  unsigned ldsOut = (unsigned)(size_t)&sOut[0];
  unsigned ldsB   = (unsigned)(size_t)&sB[0];

  // --- async global -> LDS: own row into sOut, perm row into sB -------------
  // 16 bytes/lane/op; 256 threads x 4 ops = 16KB per row.
#pragma unroll
  for (int v = 0; v < 4; ++v) {
    unsigned byteOff = (unsigned)(v * NT + tid) * 16u;
    unsigned la = ldsOut + byteOff;
    unsigned ga = rowABytes + byteOff;
    asm volatile("global_load_async_to_lds_b128 %0, %1, %2"
                 :: "v"(la), "v"(ga), "s"(x) : "memory");
    unsigned lb = ldsB + byteOff;
    unsigned gb = rowBBytes + byteOff;
    asm volatile("global_load_async_to_lds_b128 %0, %1, %2"
                 :: "v"(lb), "v"(gb), "s"(x) : "memory");
  }
  asm volatile("s_wait_asynccnt 0x0" ::: "memory");
  __syncthreads();

  // --- build sortable keys --------------------------------------------------
#pragma unroll
  for (int v = 0; v < N_ELEM / NT; ++v) {
    int e = v * NT + tid;
    unsigned ub = __float_as_uint(sOut[e]);
    sA[e] = (((unsigned long long)b2k(ub)) << 32) | (unsigned)e;
    sB[e] = b2k(sB[e]);
  }
  __syncthreads();

  // --- joint bitonic sort of sA (64b composite) and sB (32b keys) -----------
  for (unsigned k = 2; k <= N_ELEM; k <<= 1) {
    for (unsigned j = k >> 1; j > 0; j >>= 1) {
      // 2048 pairs in each array; 256 threads -> 8 slices per array.
#pragma unroll
      for (int v = 0; v < (N_ELEM / 2) / NT; ++v) {
        unsigned p = (unsigned)(v * NT + tid);
        unsigned i = ((p & ~(j - 1u)) << 1) | (p & (j - 1u));
        unsigned ixj = i | j;
        bool up = ((i & k) == 0u);
        unsigned long long a0 = sA[i], a1 = sA[ixj];
        if ((a0 > a1) == up) { sA[i] = a1; sA[ixj] = a0; }
      }
#pragma unroll
      for (int v = 0; v < (N_ELEM / 2) / NT; ++v) {
        unsigned p = (unsigned)(v * NT + tid);
        unsigned i = ((p & ~(j - 1u)) << 1) | (p & (j - 1u));
        unsigned ixj = i | j;
        bool up = ((i & k) == 0u);
        unsigned b0 = sB[i], b1 = sB[ixj];
        if ((b0 > b1) == up) { sB[i] = b1; sB[ixj] = b0; }
      }
      __syncthreads();
    }
  }

  // --- blend + scatter through carried index --------------------------------
  const float w = 1.0f - lm;
#pragma unroll
  for (int v = 0; v < N_ELEM / NT; ++v) {
    int jj = v * NT + tid;
    unsigned long long comp = sA[jj];
    unsigned i = (unsigned)comp;                 // original index (rank jj)
    float val  = k2f((unsigned)(comp >> 32));    // own value
    float pval = k2f(sB[jj]);                    // perm-row sorted value at same rank
    sOut[i] = val + (pval - val) * w;
  }
  __syncthreads();

  // --- async LDS -> global writeback (no VGPR bounce) -----------------------
#pragma unroll
  for (int v = 0; v < 4; ++v) {
    unsigned byteOff = (unsigned)(v * NT + tid) * 16u;
    unsigned l = ldsOut + byteOff;
    unsigned g = rowABytes + byteOff;
    asm volatile("global_store_async_from_lds_b128 %0, %1, %2"
                 :: "v"(g), "v"(l), "s"(out) : "memory");
  }
  asm volatile("s_wait_asynccnt 0x0" ::: "memory");
}

extern "C" void kernel_launch(void* const* d_in, const int* in_sizes, int n_in,
                              void* d_out, int out_size, void* d_ws, size_t ws_size,
                              hipStream_t stream) {
  (void)in_sizes; (void)n_in; (void)out_size; (void)d_ws; (void)ws_size;
  const float* x       = (const float*)d_in[0];
  const float* lmda    = (const float*)d_in[1];
  const long long* prm = (const long long*)d_in[2];
  float* out           = (float*)d_out;

  dim3 grid(B_DIM * C_DIM);  // 8192 rows, one workgroup each
  dim3 block(NT);            // 8 wave32s
  efdmix_kernel<<<grid, block, 0, stream>>>(x, lmda, prm, out);
}